// DirectVolumeRendering_47691316854847
// MI455X (gfx1250) — compile-verified
//
#include <hip/hip_runtime.h>

typedef __attribute__((ext_vector_type(2))) float v2f;
typedef __attribute__((ext_vector_type(8))) float v8f;

#define RES   128
#define NS    64          // ray samples
#define NF    16          // volume feature channels
#define GRID  96
#define CF    32
#define NRAY  (RES * RES)
#define VOLC  (GRID * GRID * GRID)   // 884736 elems per channel
#define WAVES 8                      // waves (rays) per block
#define LPAD  18                     // padded LDS row stride (even -> 8B aligned float2)

// Fused direct-volume-rendering kernel.
// One wave32 == one ray. Trilinear gather (memory-bound, L2-resident volume),
// dens@Wf via V_WMMA_F32_16X16X4_F32 (full f32 precision), NeRF compositing.
__global__ __launch_bounds__(256) void dvr_kernel(
    const float* __restrict__ volume,     // [B,16,96,96,96]
    const float* __restrict__ world_mat,  // [B,4,4]
    const float* __restrict__ camera_mat, // [1,4,4]
    const float* __restrict__ Wf,         // [16,32]
    const float* __restrict__ bf,         // [32]
    const float* __restrict__ Wa,         // [16,1]
    const float* __restrict__ ba,         // [1]
    float* __restrict__ out)              // [B,32,128,128]
{
    __shared__ float sdens[WAVES][NS][LPAD];  // 36864 B
    __shared__ float salpha[WAVES][NS];       //  2048 B
    __shared__ float sw[WAVES][NS];           //  2048 B

    const int tid  = threadIdx.x;
    const int wv   = tid >> 5;
    const int lane = tid & 31;
    const int m    = lane & 15;   // N / M-within-half index
    const int h    = lane >> 4;   // lane half (K/M offset selector)

    const int ray = blockIdx.x * WAVES + wv;   // grid sized exactly B*NRAY/WAVES
    const int b   = ray / NRAY;
    const int n   = ray % NRAY;
    const int gx  = n / RES;
    const int gy  = n % RES;

    // ---- camera: M = world_mat @ camera_mat (rows 0..2 only) ----
    const float* W = world_mat + b * 16;
    float M[3][4];
#pragma unroll
    for (int i = 0; i < 3; ++i)
#pragma unroll
        for (int j = 0; j < 4; ++j) {
            float a = 0.f;
#pragma unroll
            for (int k = 0; k < 4; ++k) a += W[i * 4 + k] * camera_mat[k * 4 + j];
            M[i][j] = a;
        }
    const float px = 2.f * (float)gx / (float)(RES - 1) - 1.f;
    const float py = -(2.f * (float)gy / (float)(RES - 1) - 1.f);
    float cam[3], dir[3];
#pragma unroll
    for (int i = 0; i < 3; ++i) {
        const float pw = M[i][0] * px + M[i][1] * py - M[i][2] + M[i][3];
        cam[i] = M[i][3];
        dir[i] = pw - cam[i];
    }

    // ---- small constants into registers ----
    float wa[NF];
#pragma unroll
    for (int f = 0; f < NF; ++f) wa[f] = Wa[f];
    const float bav   = ba[0];
    const float bias0 = bf[m];
    const float bias1 = bf[16 + m];

    // B fragments for V_WMMA_F32_16X16X4_F32:
    // lane L holds B[K = 4j + 2h + {0,1}][N = 16*nt + m]
    v2f Bf[2][4];
#pragma unroll
    for (int nt = 0; nt < 2; ++nt)
#pragma unroll
        for (int j = 0; j < 4; ++j) {
            Bf[nt][j].x = Wf[(4 * j + 2 * h + 0) * CF + 16 * nt + m];
            Bf[nt][j].y = Wf[(4 * j + 2 * h + 1) * CF + 16 * nt + m];
        }

    const float* volB = volume + (size_t)b * NF * VOLC;

    // ---- phase 1: trilinear sampling, alphas, stage dens to LDS ----
#pragma unroll
    for (int half = 0; half < 2; ++half) {
        const int   s  = lane + 32 * half;
        const float di = 1.f + (2.f / 63.f) * (float)s;
        float pt[3];
#pragma unroll
        for (int i = 0; i < 3; ++i) pt[i] = cam[i] + di * dir[i];

        const bool valid = pt[0] <= 1.01f && pt[0] >= -1.01f &&
                           pt[1] <= 1.01f && pt[1] >= -1.01f &&
                           pt[2] <= 1.01f && pt[2] >= -1.01f;

        const float fx = fminf(fmaxf((pt[0] + 1.f) * 0.5f * (GRID - 1), 0.f), (float)(GRID - 1));
        const float fy = fminf(fmaxf((pt[1] + 1.f) * 0.5f * (GRID - 1), 0.f), (float)(GRID - 1));
        const float fz = fminf(fmaxf((pt[2] + 1.f) * 0.5f * (GRID - 1), 0.f), (float)(GRID - 1));
        int x0 = (int)fx; if (x0 > GRID - 2) x0 = GRID - 2;
        int y0 = (int)fy; if (y0 > GRID - 2) y0 = GRID - 2;
        int z0 = (int)fz; if (z0 > GRID - 2) z0 = GRID - 2;
        const float wx = fx - (float)x0, wy = fy - (float)y0, wz = fz - (float)z0;
        const float ix = 1.f - wx, iy = 1.f - wy, iz = 1.f - wz;

        const float w000 = iz * iy * ix, w001 = iz * iy * wx;
        const float w010 = iz * wy * ix, w011 = iz * wy * wx;
        const float w100 = wz * iy * ix, w101 = wz * iy * wx;
        const float w110 = wz * wy * ix, w111 = wz * wy * wx;

        const int base = (z0 * GRID + y0) * GRID + x0;
        const float vmask = valid ? 1.f : 0.f;

        float zacc = bav;
        float d[NF];
#pragma unroll
        for (int f = 0; f < NF; ++f) {
            const float* p = volB + (size_t)f * VOLC + base;
            const float v000 = p[0];
            const float v001 = p[1];
            const float v010 = p[GRID];
            const float v011 = p[GRID + 1];
            const float v100 = p[GRID * GRID];
            const float v101 = p[GRID * GRID + 1];
            const float v110 = p[GRID * GRID + GRID];
            const float v111 = p[GRID * GRID + GRID + 1];
            float dv = w000 * v000 + w001 * v001 + w010 * v010 + w011 * v011 +
                       w100 * v100 + w101 * v101 + w110 * v110 + w111 * v111;
            dv *= vmask;
            d[f]  = dv;
            zacc += dv * wa[f];
        }
        salpha[wv][s] = 1.f / (1.f + __expf(-zacc));
#pragma unroll
        for (int f = 0; f < NF; f += 2) {
            *(v2f*)&sdens[wv][s][f] = (v2f){d[f], d[f + 1]};
        }
    }
    __syncthreads();

    // ---- phase 2: serial transmittance scan (64 mults, one lane per ray) ----
    if (lane == 0) {
        float T = 1.f;
        for (int s = 0; s < NS; ++s) {
            const float a = salpha[wv][s];
            sw[wv][s] = a * T;
            T *= (1.f - a + 1e-10f);
        }
    }
    __syncthreads();

    // ---- phase 3: WMMA (f32, 16x16x4) + bias/ReLU + composite accumulation ----
    float acc0 = 0.f, acc1 = 0.f;
#pragma unroll
    for (int t = 0; t < 4; ++t) {              // M-tiles over 64 samples
        v8f c0 = {};
        v8f c1 = {};
#pragma unroll
        for (int j = 0; j < 4; ++j) {          // K-steps over 16 features
            const v2f a = *(const v2f*)&sdens[wv][16 * t + m][4 * j + 2 * h];
            c0 = __builtin_amdgcn_wmma_f32_16x16x4_f32(false, a, false, Bf[0][j],
                                                       (short)0, c0, false, false);
            c1 = __builtin_amdgcn_wmma_f32_16x16x4_f32(false, a, false, Bf[1][j],
                                                       (short)0, c1, false, false);
        }
#pragma unroll
        for (int v = 0; v < 8; ++v) {          // C/D row v -> sample M = 16t + 8h + v
            const float wgt = sw[wv][16 * t + 8 * h + v];
            acc0 += wgt * fmaxf(c0[v] + bias0, 0.f);
            acc1 += wgt * fmaxf(c1[v] + bias1, 0.f);
        }
    }

    // cross-half reduction: lanes L and L^16 hold complementary M-halves
    acc0 += __shfl_xor(acc0, 16, 32);
    acc1 += __shfl_xor(acc1, 16, 32);

    // img[b][c][gy][gx] = feat[b][n][c]; lanes 0-15 -> c=m (acc0), 16-31 -> c=16+m (acc1)
    const int   c   = lane;
    const float val = (lane < 16) ? acc0 : acc1;
    out[(((size_t)b * CF + c) * RES + gy) * RES + gx] = val;
}

extern "C" void kernel_launch(void* const* d_in, const int* in_sizes, int n_in,
                              void* d_out, int out_size, void* d_ws, size_t ws_size,
                              hipStream_t stream) {
    const float* volume     = (const float*)d_in[0];
    const float* world_mat  = (const float*)d_in[1];
    const float* camera_mat = (const float*)d_in[2];
    const float* Wf         = (const float*)d_in[3];
    const float* bf         = (const float*)d_in[4];
    const float* Wa         = (const float*)d_in[5];
    const float* ba         = (const float*)d_in[6];
    float* out = (float*)d_out;

    const int B = 4;
    const int nblocks = (B * NRAY) / WAVES;   // 8192
    dvr_kernel<<<nblocks, 256, 0, stream>>>(volume, world_mat, camera_mat,
                                            Wf, bf, Wa, ba, out);
}